// MambaTorchModel_77421080477848
// MI455X (gfx1250) — compile-verified
//
#include <hip/hip_runtime.h>
#include <math.h>

typedef __attribute__((ext_vector_type(2))) float v2f;
typedef __attribute__((ext_vector_type(8))) float v8f;

constexpr int cB = 8, cT = 512, cNM = 128, cDM = 512, cDI = 1024;
constexpr int cDS = 32, cDR = 32, cDC = 4, cNC = 100;
constexpr int cM = cB * cT;                 // 4096 rows
constexpr int cXD = cDR + 2 * cDS;          // 96

__device__ __forceinline__ float siluf(float x) { return x / (1.f + __expf(-x)); }
__device__ __forceinline__ float softplusf(float x) { return x > 20.f ? x : log1pf(__expf(x)); }

__device__ __forceinline__ v8f wmma_f32(v2f a, v2f b, v8f c) {
  return __builtin_amdgcn_wmma_f32_16x16x4_f32(false, a, false, b, (short)0, c,
                                               false, false);
}

// ---------------------------------------------------------------------------
// Async-staged FP32 WMMA GEMM.
// Block = 8 waves = 256 rows x one 64-col strip. The B K-panel (32x64) is
// double-buffered in LDS via GLOBAL_LOAD_ASYNC_TO_LDS_B128 (2 transfers per
// thread per panel), synchronized with s_wait_asynccnt + barrier. Waves read
// B fragments from LDS (row stride 80 floats -> the two half-wave windows,
// 2 rows apart, hit disjoint bank groups). The full panel of A fragments
// (16 x global_load_b64) is hoisted into registers BEFORE the async-wait +
// barrier so their latency overlaps the panel sync instead of stalling each
// WMMA group on s_wait_loadcnt 0.
// Requires: M % 256 == 0, N % 64 == 0, K % 32 == 0.
// EPI: 0 none, 1 +bias, 2 softplus(x+bias), 3 +resid
// ---------------------------------------------------------------------------
constexpr int KC = 32;          // K-panel depth
constexpr int KSTEPS = KC / 4;  // 8 WMMA k-steps per panel
constexpr int BROW = 80;        // padded LDS row stride in floats

template <int EPI>
__global__ __launch_bounds__(256) void gemm_wmma_async(
    const float* __restrict__ A, int lda,
    const float* __restrict__ Bm,
    const float* __restrict__ bias,
    const float* __restrict__ resid,
    float* __restrict__ C,
    int M, int N, int K) {
  __shared__ float sB[2][KC * BROW];

  const int nstrips = N / 64;
  const int mblk = blockIdx.x / nstrips;
  const int n0 = (blockIdx.x % nstrips) * 64;
  const int wave = threadIdx.x >> 5;
  const int lane = threadIdx.x & 31;
  const int m0 = mblk * 256 + wave * 32;

  const int mA = lane & 15;
  const int kh = (lane >> 4) << 1;         // 0 or 2
  const int ln = lane & 15;

  // panel fill: thread t moves rows fk and fk+16, 4 floats at column fc
  const int fc = (threadIdx.x & 15) * 4;
  const int fk = threadIdx.x >> 4;         // 0..15
  auto fill = [&](int buf, int kc0) {
    const float* g0 = Bm + (size_t)(kc0 + fk) * N + n0 + fc;
    const float* g1 = g0 + (size_t)16 * N;
    unsigned l0 = (unsigned)(size_t)&sB[buf][fk * BROW + fc];
    unsigned l1 = (unsigned)(size_t)&sB[buf][(fk + 16) * BROW + fc];
    asm volatile("global_load_async_to_lds_b128 %0, %1, off"
                 :: "v"(l0), "v"(g0) : "memory");
    asm volatile("global_load_async_to_lds_b128 %0, %1, off"
                 :: "v"(l1), "v"(g1) : "memory");
  };

  v8f zero = {};
  v8f acc[2][4];
#pragma unroll
  for (int mi = 0; mi < 2; ++mi)
#pragma unroll
    for (int i = 0; i < 4; ++i) acc[mi][i] = zero;

  const float* Ab0 = A + (size_t)(m0 + mA) * lda + kh;
  const float* Ab1 = A + (size_t)(m0 + 16 + mA) * lda + kh;

  fill(0, 0);
  int cur = 0;
  for (int kc = 0; kc < K; kc += KC) {
    // ---- hoist the whole panel of A fragments (overlaps async wait+barrier)
    v2f a0[KSTEPS], a1[KSTEPS];
#pragma unroll
    for (int j = 0; j < KSTEPS; ++j) {
      a0[j] = *(const v2f*)(Ab0 + kc + 4 * j);     // global_load_b64
      a1[j] = *(const v2f*)(Ab1 + kc + 4 * j);
    }

    if (kc + KC < K) {
      fill(cur ^ 1, kc + KC);                      // prefetch next panel
      asm volatile("s_wait_asynccnt 0x2" ::: "memory");  // current panel done
    } else {
      asm volatile("s_wait_asynccnt 0x0" ::: "memory");
    }
    __syncthreads();

    const float* sCur = &sB[cur][0];
#pragma unroll
    for (int j = 0; j < KSTEPS; ++j) {
      const int k0 = 4 * j;
      const float* sp = sCur + (k0 + kh) * BROW + ln;
      v2f b[4];
#pragma unroll
      for (int i = 0; i < 4; ++i) {
        b[i].x = sp[i * 16];
        b[i].y = sp[BROW + i * 16];
      }
#pragma unroll
      for (int i = 0; i < 4; ++i) {
        acc[0][i] = wmma_f32(a0[j], b[i], acc[0][i]);
        acc[1][i] = wmma_f32(a1[j], b[i], acc[1][i]);
      }
    }
    __syncthreads();                               // before next overwrite
    cur ^= 1;
  }

  const int cbase = n0 + ln;
#pragma unroll
  for (int mi = 0; mi < 2; ++mi) {
    const int rbase = m0 + mi * 16 + ((lane >> 4) << 3);
#pragma unroll
    for (int i = 0; i < 4; ++i) {
#pragma unroll
      for (int r = 0; r < 8; ++r) {
        const int rr = rbase + r;
        const int cc = cbase + i * 16;
        float v = acc[mi][i][r];
        if (EPI == 1 || EPI == 2) v += bias[cc];
        if (EPI == 2) v = softplusf(v);
        if (EPI == 3) v += resid[(size_t)rr * N + cc];
        C[(size_t)rr * N + cc] = v;
      }
    }
  }
}

// ---------------------------------------------------------------------------
// Direct-load WMMA GEMM (for narrow N, e.g. N=96). 32x32 tile per wave.
// ---------------------------------------------------------------------------
template <int NT, int EPI>
__global__ void gemm_wmma(const float* __restrict__ A, int lda,
                          const float* __restrict__ Bm,
                          const float* __restrict__ bias,
                          const float* __restrict__ resid,
                          float* __restrict__ C,
                          int M, int N, int K) {
  const int wave = (blockIdx.x * blockDim.x + threadIdx.x) >> 5;
  const int lane = threadIdx.x & 31;
  const int nstrips = N / (16 * NT);
  const int m0 = (wave / nstrips) * 32;
  const int n0 = (wave % nstrips) * (16 * NT);
  if (m0 >= M) return;                     // wave-uniform

  const int mA = lane & 15;
  const int kh = (lane >> 4) << 1;

  v8f zero = {};
  v8f acc[2][NT];
#pragma unroll
  for (int mi = 0; mi < 2; ++mi)
#pragma unroll
    for (int i = 0; i < NT; ++i) acc[mi][i] = zero;

  const float* Ab0 = A + (size_t)(m0 + mA) * lda + kh;
  const float* Ab1 = A + (size_t)(m0 + 16 + mA) * lda + kh;
  const float* Bb = Bm + (size_t)kh * N + n0 + (lane & 15);

  for (int k0 = 0; k0 < K; k0 += 4) {
    v2f a[2];
    a[0] = *(const v2f*)(Ab0 + k0);
    a[1] = *(const v2f*)(Ab1 + k0);
    const float* Bp = Bb + (size_t)k0 * N;
    v2f b[NT];
#pragma unroll
    for (int i = 0; i < NT; ++i) {
      b[i].x = Bp[i * 16];
      b[i].y = Bp[i * 16 + N];
    }
#pragma unroll
    for (int mi = 0; mi < 2; ++mi)
#pragma unroll
      for (int i = 0; i < NT; ++i)
        acc[mi][i] = wmma_f32(a[mi], b[i], acc[mi][i]);
  }

  const int cbase = n0 + (lane & 15);
#pragma unroll
  for (int mi = 0; mi < 2; ++mi) {
    const int rbase = m0 + mi * 16 + ((lane >> 4) << 3);
#pragma unroll
    for (int i = 0; i < NT; ++i) {
#pragma unroll
      for (int r = 0; r < 8; ++r) {
        const int rr = rbase + r;
        const int cc = cbase + i * 16;
        float v = acc[mi][i][r];
        if (EPI == 1 || EPI == 2) v += bias[cc];
        if (EPI == 2) v = softplusf(v);
        if (EPI == 3) v += resid[(size_t)rr * N + cc];
        C[(size_t)rr * N + cc] = v;
      }
    }
  }
}

// RMSNorm over DM per row
__global__ void rmsnorm_k(const float* __restrict__ h0, const float* __restrict__ w,
                          float* __restrict__ xn) {
  __shared__ float red[256];
  const int row = blockIdx.x;
  const float* hr = h0 + (size_t)row * cDM;
  float s = 0.f;
  for (int c = threadIdx.x; c < cDM; c += 256) { float v = hr[c]; s += v * v; }
  red[threadIdx.x] = s;
  __syncthreads();
  for (int off = 128; off > 0; off >>= 1) {
    if (threadIdx.x < off) red[threadIdx.x] += red[threadIdx.x + off];
    __syncthreads();
  }
  const float scale = rsqrtf(red[0] / (float)cDM + 1e-5f);
  for (int c = threadIdx.x; c < cDM; c += 256)
    xn[(size_t)row * cDM + c] = hr[c] * scale * w[c];
}

// causal depthwise conv1d (left pad DC-1) + bias + SiLU
__global__ void conv_silu_k(const float* __restrict__ xr, const float* __restrict__ cw,
                            const float* __restrict__ cb, float* __restrict__ xp) {
  const int idx = blockIdx.x * blockDim.x + threadIdx.x;
  if (idx >= cM * cDI) return;
  const int d = idx % cDI;
  const int row = idx / cDI;
  const int t = row % cT;
  const int b = row / cT;
  float acc = cb[d];
#pragma unroll
  for (int j = 0; j < cDC; ++j) {
    const int tt = t - (cDC - 1) + j;
    if (tt >= 0)
      acc += xr[((size_t)(b * cT + tt)) * (2 * cDI) + d] * cw[d * cDC + j];
  }
  xp[(size_t)row * cDI + d] = siluf(acc);
}

// selective scan: state h[DS] in VGPRs, per-timestep B_t/C_t via LDS broadcast
__global__ void scan_k(const float* __restrict__ xdbl, const float* __restrict__ delta,
                       const float* __restrict__ xp, const float* __restrict__ xr,
                       const float* __restrict__ A_log, const float* __restrict__ Dv,
                       float* __restrict__ yg) {
  __shared__ float sB[cDS];
  __shared__ float sC[cDS];
  const int chunks = cDI / 256;
  const int chunk = blockIdx.x % chunks;
  const int b = blockIdx.x / chunks;
  const int d = chunk * 256 + threadIdx.x;

  float Arow[cDS];
#pragma unroll
  for (int s = 0; s < cDS; ++s) Arow[s] = -__expf(A_log[(size_t)d * cDS + s]);
  const float Dd = Dv[d];

  float h[cDS];
#pragma unroll
  for (int s = 0; s < cDS; ++s) h[s] = 0.f;

  for (int t = 0; t < cT; ++t) {
    const size_t row = (size_t)b * cT + t;
    if (threadIdx.x < cDS)
      sB[threadIdx.x] = xdbl[row * cXD + cDR + threadIdx.x];
    else if (threadIdx.x < 2 * cDS)
      sC[threadIdx.x - cDS] = xdbl[row * cXD + cDR + cDS + (threadIdx.x - cDS)];
    __syncthreads();

    const float dt = delta[row * cDI + d];
    const float u = xp[row * cDI + d];
    const float du = dt * u;
    float y = 0.f;
#pragma unroll
    for (int s = 0; s < cDS; ++s) {
      h[s] = __expf(dt * Arow[s]) * h[s] + du * sB[s];
      y += h[s] * sC[s];
    }
    const float res = xr[row * (2 * cDI) + cDI + d];
    yg[row * cDI + d] = (y + u * Dd) * siluf(res);
    __syncthreads();
  }
}

// pooled[row] = mean over DM columns of enc
__global__ void rowmean_k(const float* __restrict__ enc, float* __restrict__ pooled) {
  __shared__ float red[256];
  const int row = blockIdx.x;
  float s = 0.f;
  for (int c = threadIdx.x; c < cDM; c += 256) s += enc[(size_t)row * cDM + c];
  red[threadIdx.x] = s;
  __syncthreads();
  for (int off = 128; off > 0; off >>= 1) {
    if (threadIdx.x < off) red[threadIdx.x] += red[threadIdx.x + off];
    __syncthreads();
  }
  if (threadIdx.x == 0) pooled[row] = red[0] / (float)cDM;
}

// out[b,c] = pooled[b,:] @ w_cls + b_cls  (8x100, tiny)
__global__ void cls_k(const float* __restrict__ pooled, const float* __restrict__ wc,
                      const float* __restrict__ bc, float* __restrict__ out) {
  const int idx = blockIdx.x * blockDim.x + threadIdx.x;
  if (idx >= cB * cNC) return;
  const int b = idx / cNC, c = idx % cNC;
  float s = bc[c];
  for (int t = 0; t < cT; ++t) s += pooled[b * cT + t] * wc[t * cNC + c];
  out[idx] = s;
}

extern "C" void kernel_launch(void* const* d_in, const int* in_sizes, int n_in,
                              void* d_out, int out_size, void* d_ws, size_t ws_size,
                              hipStream_t stream) {
  const float* x      = (const float*)d_in[0];
  const float* w_proj = (const float*)d_in[1];
  const float* b_proj = (const float*)d_in[2];
  const float* rms_w  = (const float*)d_in[3];
  const float* w_in   = (const float*)d_in[4];
  const float* conv_w = (const float*)d_in[5];
  const float* conv_b = (const float*)d_in[6];
  const float* w_xproj= (const float*)d_in[7];
  const float* w_dt   = (const float*)d_in[8];
  const float* b_dt   = (const float*)d_in[9];
  const float* A_log  = (const float*)d_in[10];
  const float* Dv     = (const float*)d_in[11];
  const float* w_out  = (const float*)d_in[12];
  const float* w_cls  = (const float*)d_in[13];
  const float* b_cls  = (const float*)d_in[14];
  float* out = (float*)d_out;

  float* ws = (float*)d_ws;
  size_t off = 0;
  float* h0     = ws + off; off += (size_t)cM * cDM;        // 4096x512
  float* xn     = ws + off; off += (size_t)cM * cDM;        // 4096x512
  float* xr     = ws + off; off += (size_t)cM * 2 * cDI;    // 4096x2048
  float* xp     = ws + off; off += (size_t)cM * cDI;        // 4096x1024
  float* xdbl   = ws + off; off += (size_t)cM * cXD;        // 4096x96
  float* delta  = ws + off; off += (size_t)cM * cDI;        // 4096x1024
  float* yg     = ws + off; off += (size_t)cM * cDI;        // 4096x1024
  float* enc    = ws + off; off += (size_t)cM * cDM;        // 4096x512
  float* pooled = ws + off; off += (size_t)cM;              // 4096

  // 1) h0 = x @ w_proj + b_proj        (M=4096, K=128, N=512)
  gemm_wmma_async<1><<<(cM / 256) * (cDM / 64), 256, 0, stream>>>(
      x, cNM, w_proj, b_proj, nullptr, h0, cM, cDM, cNM);

  // 2) RMSNorm
  rmsnorm_k<<<cM, 256, 0, stream>>>(h0, rms_w, xn);

  // 3) xr = xn @ w_in                  (M=4096, K=512, N=2048)
  gemm_wmma_async<0><<<(cM / 256) * ((2 * cDI) / 64), 256, 0, stream>>>(
      xn, cDM, w_in, nullptr, nullptr, xr, cM, 2 * cDI, cDM);

  // 4) causal depthwise conv + SiLU -> xp
  conv_silu_k<<<(cM * cDI + 255) / 256, 256, 0, stream>>>(xr, conv_w, conv_b, xp);

  // 5) xdbl = xp @ w_xproj             (M=4096, K=1024, N=96) -- narrow N
  gemm_wmma<2, 0><<<(int)(((long)(cM / 32) * (cXD / 32) * 32 + 255) / 256), 256, 0,
                    stream>>>(xp, cDI, w_xproj, nullptr, nullptr, xdbl, cM, cXD, cDI);

  // 6) delta = softplus(dt @ w_dt + b_dt)   (A = xdbl[:, :32], lda=96, K=32)
  gemm_wmma_async<2><<<(cM / 256) * (cDI / 64), 256, 0, stream>>>(
      xdbl, cXD, w_dt, b_dt, nullptr, delta, cM, cDI, cDR);

  // 7) selective scan + D-skip + gate -> yg
  scan_k<<<cB * (cDI / 256), 256, 0, stream>>>(xdbl, delta, xp, xr, A_log, Dv, yg);

  // 8) enc = h0 + yg @ w_out           (M=4096, K=1024, N=512)
  gemm_wmma_async<3><<<(cM / 256) * (cDM / 64), 256, 0, stream>>>(
      yg, cDI, w_out, nullptr, h0, enc, cM, cDM, cDI);

  // 9) pooled = mean(enc, axis=feature)
  rowmean_k<<<cM, 256, 0, stream>>>(enc, pooled);

  // 10) out = pooled @ w_cls + b_cls
  cls_k<<<(cB * cNC + 255) / 256, 256, 0, stream>>>(pooled, w_cls, b_cls, out);
}